// G_Unet_79216376807798
// MI455X (gfx1250) — compile-verified
//
#include <hip/hip_runtime.h>
#include <hip/hip_bf16.h>
#include <math.h>

// ---------------------------------------------------------------------------
// GraphUNet (N=4096, DEPTH=3, HID=32) for MI455X / gfx1250.
// Heavy compute = dense symmetric A@A (augment) -> f32 WMMA 16x16x4 with
// double-buffered async global->LDS staging (ASYNCcnt path).
// GCN propagation (n x n @ n x 32) also on WMMA.
// ---------------------------------------------------------------------------

#define NNODES 4096
#define HID 32

typedef __attribute__((ext_vector_type(2))) float v2f;
typedef __attribute__((ext_vector_type(8))) float v8f;

#if defined(__has_builtin)
#if __has_builtin(__builtin_amdgcn_global_load_async_to_lds_b128)
#define HAVE_ASYNC_LDS 1
#endif
#endif
#ifndef HAVE_ASYNC_LDS
#define HAVE_ASYNC_LDS 0
#endif

#if HAVE_ASYNC_LDS
// builtin signature (from hipcc diagnostic): 
//   void (v4i addrspace(1)*, v4i addrspace(3)*, imm int offset, imm int cpol)
typedef int v4i_vs __attribute__((vector_size(4 * sizeof(int))));
typedef __attribute__((address_space(1))) v4i_vs* gptr_b128;
typedef __attribute__((address_space(3))) v4i_vs* lptr_b128;
#if __has_builtin(__builtin_amdgcn_s_wait_asynccnt)
#define WAIT_ASYNC(n) __builtin_amdgcn_s_wait_asynccnt(n)
#else
#define WAIT_ASYNC(n) asm volatile("s_wait_asynccnt %0" ::"i"(n) : "memory")
#endif
#else
#define WAIT_ASYNC(n) ((void)0)
#endif

__device__ inline v8f wmma_f32_16x16x4(v2f a, v2f b, v8f c) {
    // 8 args: (neg_a, A, neg_b, B, c_mod, C, reuse_a, reuse_b)
    return __builtin_amdgcn_wmma_f32_16x16x4_f32(false, a, false, b, (short)0, c,
                                                 false, false);
}

__device__ inline float gelu_exact(float v) {
    return 0.5f * v * (1.0f + erff(v * 0.7071067811865476f));
}

// ------------------------- adjacency build ---------------------------------
__global__ void build_adj(const int* ei, float* A, int E, int n) {
    int e = blockIdx.x * blockDim.x + threadIdx.x;
    if (e >= E) return;
    int i = ei[e];
    int j = ei[E + e];
    if (i == j) return;
    A[(size_t)i * n + j] = 1.0f;  // benign same-value races
    A[(size_t)j * n + i] = 1.0f;
}

// ----------------- degree (of A + 2I), dinv = rsqrt(deg) -------------------
__global__ void degree_rsqrt(const float* A, float* dinv, int n) {
    __shared__ float red[256];
    int i = blockIdx.x;
    float s = 0.0f;
    for (int j = threadIdx.x; j < n; j += 256) s += A[(size_t)i * n + j];
    red[threadIdx.x] = s;
    __syncthreads();
    for (int off = 128; off > 0; off >>= 1) {
        if (threadIdx.x < off) red[threadIdx.x] += red[threadIdx.x + off];
        __syncthreads();
    }
    if (threadIdx.x == 0) dinv[i] = rsqrtf(red[0] + 2.0f);
}

// --------------------- Yp = dinv[i] * (x @ W),  W: F x 32 ------------------
__global__ void xw_scale(const float* x, const float* W, const float* dinv,
                         float* Yp, int n, int F) {
    int idx = blockIdx.x * blockDim.x + threadIdx.x;
    if (idx >= n * HID) return;
    int i = idx >> 5, h = idx & 31;
    float s = 0.0f;
    for (int f = 0; f < F; ++f) s += x[(size_t)i * F + f] * W[f * HID + h];
    Yp[idx] = dinv[i] * s;
}

// final layer: W is 32x1 -> y[i] = dinv[i] * (x[i,:] . W)
__global__ void xw_scale_vec(const float* x, const float* W, const float* dinv,
                             float* y, int n) {
    int i = blockIdx.x * blockDim.x + threadIdx.x;
    if (i >= n) return;
    float s = 0.0f;
    for (int f = 0; f < HID; ++f) s += x[(size_t)i * HID + f] * W[f];
    y[i] = dinv[i] * s;
}

// --------- GCN propagate via WMMA:  out = dinv*(A@Yp + 2*Yp) + b -----------
// Block = 128 threads = 4 waves. Wave w: rows (w&1)*16, cols (w>>1)*16 of a
// 32-row x 32-col block tile. Grid = n/32. EXEC is always full (no divergence).
__global__ void gcn_prop(const float* __restrict__ A, const float* __restrict__ Yp,
                         const float* __restrict__ dinv, const float* __restrict__ b,
                         float* __restrict__ out, int n, int do_gelu) {
    int tid = threadIdx.x;
    int w = tid >> 5, lane = tid & 31;
    int half = lane >> 4, lanelow = lane & 15;
    int mbase = blockIdx.x * 32 + (w & 1) * 16;
    int nbase = (w >> 1) * 16;
    v8f acc = {0.f, 0.f, 0.f, 0.f, 0.f, 0.f, 0.f, 0.f};
    const float* arow = A + (size_t)(mbase + lanelow) * n;
    for (int kk = 0; kk < n; kk += 4) {
        int ka = kk + half * 2;
        v2f a;
        a[0] = arow[ka];
        a[1] = arow[ka + 1];
        v2f bf;
        bf[0] = Yp[(size_t)ka * HID + nbase + lanelow];
        bf[1] = Yp[(size_t)(ka + 1) * HID + nbase + lanelow];
        acc = wmma_f32_16x16x4(a, bf, acc);
    }
#pragma unroll
    for (int r = 0; r < 8; ++r) {
        int gm = mbase + r + half * 8;
        int gn = nbase + lanelow;
        float v = acc[r] + 2.0f * Yp[(size_t)gm * HID + gn];
        v = dinv[gm] * v + b[gn];
        if (do_gelu) v = gelu_exact(v);
        out[(size_t)gm * HID + gn] = v;
    }
}

// -------- augment: C = A@A + 2A, zero diagonal (A symmetric) ---------------
// 256 threads = 8 waves. Block tile 64x64, K tile 32. Wave w: row tile
// (w&3)*16, col base (w>>2)*32 (two 16x16 accumulators).
// LDS tiles double-buffered; filled with global_load_async_to_lds_b128 when
// available (ASYNCcnt), synchronous float4 staging otherwise.
#define ABM 64
#define ABN 64
#define ABK 32
#define ASTR 36  // 144B rows: 16B aligned; 36*m mod 64 -> 16 distinct banks
#define BSTR 68  // 272B rows: 16B aligned

__global__ void augment_gemm(const float* __restrict__ A, float* __restrict__ C,
                             int n) {
    __shared__ __align__(16) float As[2][ABM][ASTR];
    __shared__ __align__(16) float Bs[2][ABK][BSTR];
    int tid = threadIdx.x;
    int w = tid >> 5, lane = tid & 31;
    int half = lane >> 4, lanelow = lane & 15;
    int bM = blockIdx.y * ABM;
    int bN = blockIdx.x * ABN;
    int wm = (w & 3) * 16;
    int wn = (w >> 2) * 32;
    v8f acc0 = {0.f, 0.f, 0.f, 0.f, 0.f, 0.f, 0.f, 0.f};
    v8f acc1 = {0.f, 0.f, 0.f, 0.f, 0.f, 0.f, 0.f, 0.f};

    // per-thread float4 slots: A-tile 64x32 -> 512 slots, B-tile 32x64 -> 512
    int a_r = tid >> 3, a_c = (tid & 7) << 2;    // rows a_r, a_r+32
    int b_r = tid >> 4, b_c = (tid & 15) << 2;   // rows b_r, b_r+16

    auto issue_tile = [&](int buf, int kk) {
#if HAVE_ASYNC_LDS
        __builtin_amdgcn_global_load_async_to_lds_b128(
            (gptr_b128)(A + (size_t)(bM + a_r) * n + kk + a_c),
            (lptr_b128)(&As[buf][a_r][a_c]), 0, 0);
        __builtin_amdgcn_global_load_async_to_lds_b128(
            (gptr_b128)(A + (size_t)(bM + a_r + 32) * n + kk + a_c),
            (lptr_b128)(&As[buf][a_r + 32][a_c]), 0, 0);
        __builtin_amdgcn_global_load_async_to_lds_b128(
            (gptr_b128)(A + (size_t)(kk + b_r) * n + bN + b_c),
            (lptr_b128)(&Bs[buf][b_r][b_c]), 0, 0);
        __builtin_amdgcn_global_load_async_to_lds_b128(
            (gptr_b128)(A + (size_t)(kk + b_r + 16) * n + bN + b_c),
            (lptr_b128)(&Bs[buf][b_r + 16][b_c]), 0, 0);
#else
        *(float4*)(&As[buf][a_r][a_c]) =
            *(const float4*)(A + (size_t)(bM + a_r) * n + kk + a_c);
        *(float4*)(&As[buf][a_r + 32][a_c]) =
            *(const float4*)(A + (size_t)(bM + a_r + 32) * n + kk + a_c);
        *(float4*)(&Bs[buf][b_r][b_c]) =
            *(const float4*)(A + (size_t)(kk + b_r) * n + bN + b_c);
        *(float4*)(&Bs[buf][b_r + 16][b_c]) =
            *(const float4*)(A + (size_t)(kk + b_r + 16) * n + bN + b_c);
#endif
    };

    issue_tile(0, 0);
    int buf = 0;
    for (int kk = 0; kk < n; kk += ABK) {
        if (kk + ABK < n) {
            issue_tile(buf ^ 1, kk + ABK);  // prefetch next tile (other buffer)
            WAIT_ASYNC(4);                  // in-order: current tile landed
        } else {
            WAIT_ASYNC(0);
        }
        __syncthreads();
        const float(*Asb)[ASTR] = As[buf];
        const float(*Bsb)[BSTR] = Bs[buf];
#pragma unroll
        for (int k = 0; k < ABK; k += 4) {
            int ka = k + half * 2;
            v2f a;
            a[0] = Asb[wm + lanelow][ka];
            a[1] = Asb[wm + lanelow][ka + 1];
            v2f b0v;
            b0v[0] = Bsb[ka][wn + lanelow];
            b0v[1] = Bsb[ka + 1][wn + lanelow];
            v2f b1v;
            b1v[0] = Bsb[ka][wn + 16 + lanelow];
            b1v[1] = Bsb[ka + 1][wn + 16 + lanelow];
            acc0 = wmma_f32_16x16x4(a, b0v, acc0);
            acc1 = wmma_f32_16x16x4(a, b1v, acc1);
        }
        __syncthreads();
        buf ^= 1;
    }
#pragma unroll
    for (int r = 0; r < 8; ++r) {
        int gm = bM + wm + r + half * 8;
        int gn0 = bN + wn + lanelow;
        int gn1 = gn0 + 16;
        float v0 = acc0[r] + 2.0f * A[(size_t)gm * n + gn0];
        float v1 = acc1[r] + 2.0f * A[(size_t)gm * n + gn1];
        if (gm == gn0) v0 = 0.0f;
        if (gm == gn1) v1 = 0.0f;
        C[(size_t)gm * n + gn0] = v0;
        C[(size_t)gm * n + gn1] = v1;
    }
}

// ------------------------ pooling helpers ----------------------------------
__global__ void score_tanh(const float* x, const float* p, float* s, int n) {
    int i = blockIdx.x * blockDim.x + threadIdx.x;
    if (i >= n) return;
    float nrm = 0.0f, dot = 0.0f;
    for (int h = 0; h < HID; ++h) {
        float pv = p[h];
        nrm += pv * pv;
        dot += x[(size_t)i * HID + h] * pv;
    }
    s[i] = tanhf(dot * rsqrtf(nrm));
}

// rank[i] = #{j: s[j]>s[i] or (s[j]==s[i] and j<i)}; if rank<k: perm[rank]=i.
// Matches top_k descending order with lowest-index tiebreak; perm fully covered.
__global__ void topk_rank(const float* s, int* perm, int n, int k) {
    __shared__ float sh[256];
    int i = blockIdx.x * 256 + threadIdx.x;
    float si = (i < n) ? s[i] : 0.0f;
    int rank = 0;
    for (int base = 0; base < n; base += 256) {
        int j = base + threadIdx.x;
        sh[threadIdx.x] = (j < n) ? s[j] : -1e30f;
        __syncthreads();
        for (int t = 0; t < 256; ++t) {
            int jj = base + t;
            float sj = sh[t];
            rank += ((sj > si) || (sj == si && jj < i)) ? 1 : 0;
        }
        __syncthreads();
    }
    if (i < n && rank < k) perm[rank] = i;
}

__global__ void pool_x(const float* x, const float* s, const int* perm,
                       float* xp, int k) {
    int idx = blockIdx.x * blockDim.x + threadIdx.x;
    if (idx >= k * HID) return;
    int r = idx >> 5, h = idx & 31;
    int src = perm[r];
    xp[idx] = x[(size_t)src * HID + h] * s[src];
}

__global__ void pool_adj(const float* C, const int* perm, float* Ap, int k,
                         int n) {
    int idx = blockIdx.x * blockDim.x + threadIdx.x;
    if (idx >= k * k) return;
    int r = idx / k, c = idx % k;
    Ap[idx] = C[(size_t)perm[r] * n + perm[c]];
}

// dst[perm[r], h] += xsmall[r, h]   (perm entries unique -> no atomics)
__global__ void scatter_add(float* dst, const float* xsmall, const int* perm,
                            int k) {
    int idx = blockIdx.x * blockDim.x + threadIdx.x;
    if (idx >= k * HID) return;
    int r = idx >> 5, h = idx & 31;
    dst[(size_t)perm[r] * HID + h] += xsmall[idx];
}

// final layer: z = dinv[i]*(A[i,:].y + 2*y[i]) + b0 ; out = sigmoid(z)
__global__ void gcn_prop_vec_sigmoid(const float* A, const float* y,
                                     const float* dinv, const float* b,
                                     float* out, int n) {
    __shared__ float red[256];
    int i = blockIdx.x;
    float s = 0.0f;
    for (int j = threadIdx.x; j < n; j += 256) s += A[(size_t)i * n + j] * y[j];
    red[threadIdx.x] = s;
    __syncthreads();
    for (int off = 128; off > 0; off >>= 1) {
        if (threadIdx.x < off) red[threadIdx.x] += red[threadIdx.x + off];
        __syncthreads();
    }
    if (threadIdx.x == 0) {
        float z = dinv[i] * (red[0] + 2.0f * y[i]) + b[0];
        out[i] = 1.0f / (1.0f + expf(-z));
    }
}

// ---------------------------------------------------------------------------
extern "C" void kernel_launch(void* const* d_in, const int* in_sizes, int n_in,
                              void* d_out, int out_size, void* d_ws, size_t ws_size,
                              hipStream_t stream) {
    const float* x_in = (const float*)d_in[0];
    const int* eidx = (const int*)d_in[1];
    int E = in_sizes[1] / 2;

    const float *W0, *W1, *W2, *W3, *db0, *db1, *db2, *db3;
    const float *p0, *p1, *p2, *uW0, *uW1, *uW2, *ub0, *ub1, *ub2;
    if (n_in >= 20) {  // pytree leaves passed individually
        W0 = (const float*)d_in[3];  W1 = (const float*)d_in[4];
        W2 = (const float*)d_in[5];  W3 = (const float*)d_in[6];
        db0 = (const float*)d_in[7]; db1 = (const float*)d_in[8];
        db2 = (const float*)d_in[9]; db3 = (const float*)d_in[10];
        p0 = (const float*)d_in[11]; p1 = (const float*)d_in[12];
        p2 = (const float*)d_in[13];
        uW0 = (const float*)d_in[14]; uW1 = (const float*)d_in[15];
        uW2 = (const float*)d_in[16];
        ub0 = (const float*)d_in[17]; ub1 = (const float*)d_in[18];
        ub2 = (const float*)d_in[19];
    } else {  // tuples concatenated into blobs
        const float* dw = (const float*)d_in[3];
        W0 = dw; W1 = dw + 64; W2 = dw + 64 + 1024; W3 = dw + 64 + 2048;
        const float* dbb = (const float*)d_in[4];
        db0 = dbb; db1 = dbb + 32; db2 = dbb + 64; db3 = dbb + 96;
        const float* pp = (const float*)d_in[5];
        p0 = pp; p1 = pp + 32; p2 = pp + 64;
        const float* uw = (const float*)d_in[6];
        uW0 = uw; uW1 = uw + 1024; uW2 = uw + 2048;
        const float* ubb = (const float*)d_in[7];
        ub0 = ubb; ub1 = ubb + 32; ub2 = ubb + 64;
    }

    // -------------------- workspace layout (floats) ------------------------
    float* ws = (float*)d_ws;
    size_t o = 0;
    float* A0 = ws + o;   o += (size_t)4096 * 4096;
    float* A1 = ws + o;   o += (size_t)2048 * 2048;
    float* A2 = ws + o;   o += (size_t)1024 * 1024;
    float* A3 = ws + o;   o += (size_t)512 * 512;
    float* Ctmp = ws + o; o += (size_t)4096 * 4096;
    float* xs0 = ws + o;  o += (size_t)4096 * HID;
    float* xs1 = ws + o;  o += (size_t)2048 * HID;
    float* xs2 = ws + o;  o += (size_t)1024 * HID;
    float* xa = ws + o;   o += (size_t)4096 * HID;
    float* xb = ws + o;   o += (size_t)4096 * HID;
    float* Yp = ws + o;   o += (size_t)4096 * HID;
    float* dinv = ws + o; o += 4096;
    float* score = ws + o; o += 4096;
    float* yvec = ws + o;  o += 4096;
    int* perm0 = (int*)(ws + o); o += 2048;
    int* perm1 = (int*)(ws + o); o += 1024;
    int* perm2 = (int*)(ws + o); o += 512;
    (void)ws_size;

    // -------------------- build A0 -----------------------------------------
    (void)hipMemsetAsync(A0, 0, (size_t)4096 * 4096 * sizeof(float), stream);
    build_adj<<<(E + 255) / 256, 256, 0, stream>>>(eidx, A0, E, NNODES);

    // -------------------- down gcn 0 (F=2) at n=4096 -----------------------
    degree_rsqrt<<<4096, 256, 0, stream>>>(A0, dinv, 4096);
    xw_scale<<<(4096 * HID + 255) / 256, 256, 0, stream>>>(x_in, W0, dinv, Yp, 4096, 2);
    gcn_prop<<<4096 / 32, 128, 0, stream>>>(A0, Yp, dinv, db0, xs0, 4096, 1);

    // -------------------- level 0: augment + pool --------------------------
    augment_gemm<<<dim3(4096 / ABN, 4096 / ABM), 256, 0, stream>>>(A0, Ctmp, 4096);
    score_tanh<<<4096 / 256, 256, 0, stream>>>(xs0, p0, score, 4096);
    topk_rank<<<4096 / 256, 256, 0, stream>>>(score, perm0, 4096, 2048);
    pool_x<<<(2048 * HID + 255) / 256, 256, 0, stream>>>(xs0, score, perm0, xa, 2048);
    pool_adj<<<((2048 * 2048) + 255) / 256, 256, 0, stream>>>(Ctmp, perm0, A1, 2048, 4096);

    // down gcn 1 at n=2048
    degree_rsqrt<<<2048, 256, 0, stream>>>(A1, dinv, 2048);
    xw_scale<<<(2048 * HID + 255) / 256, 256, 0, stream>>>(xa, W1, dinv, Yp, 2048, HID);
    gcn_prop<<<2048 / 32, 128, 0, stream>>>(A1, Yp, dinv, db1, xs1, 2048, 1);

    // -------------------- level 1: augment + pool --------------------------
    augment_gemm<<<dim3(2048 / ABN, 2048 / ABM), 256, 0, stream>>>(A1, Ctmp, 2048);
    score_tanh<<<2048 / 256, 256, 0, stream>>>(xs1, p1, score, 2048);
    topk_rank<<<2048 / 256, 256, 0, stream>>>(score, perm1, 2048, 1024);
    pool_x<<<(1024 * HID + 255) / 256, 256, 0, stream>>>(xs1, score, perm1, xa, 1024);
    pool_adj<<<((1024 * 1024) + 255) / 256, 256, 0, stream>>>(Ctmp, perm1, A2, 1024, 2048);

    // down gcn 2 at n=1024
    degree_rsqrt<<<1024, 256, 0, stream>>>(A2, dinv, 1024);
    xw_scale<<<(1024 * HID + 255) / 256, 256, 0, stream>>>(xa, W2, dinv, Yp, 1024, HID);
    gcn_prop<<<1024 / 32, 128, 0, stream>>>(A2, Yp, dinv, db2, xs2, 1024, 1);

    // -------------------- level 2: augment + pool --------------------------
    augment_gemm<<<dim3(1024 / ABN, 1024 / ABM), 256, 0, stream>>>(A2, Ctmp, 1024);
    score_tanh<<<1024 / 256, 256, 0, stream>>>(xs2, p2, score, 1024);
    topk_rank<<<1024 / 256, 256, 0, stream>>>(score, perm2, 1024, 512);
    pool_x<<<(512 * HID + 255) / 256, 256, 0, stream>>>(xs2, score, perm2, xa, 512);
    pool_adj<<<((512 * 512) + 255) / 256, 256, 0, stream>>>(Ctmp, perm2, A3, 512, 1024);

    // bottom gcn at n=512 -> xb
    degree_rsqrt<<<512, 256, 0, stream>>>(A3, dinv, 512);
    xw_scale<<<(512 * HID + 255) / 256, 256, 0, stream>>>(xa, W3, dinv, Yp, 512, HID);
    gcn_prop<<<512 / 32, 128, 0, stream>>>(A3, Yp, dinv, db3, xb, 512, 1);

    // -------------------- up path i=0 (j=2, n=1024) ------------------------
    (void)hipMemcpyAsync(xa, xs2, (size_t)1024 * HID * sizeof(float),
                         hipMemcpyDeviceToDevice, stream);
    scatter_add<<<(512 * HID + 255) / 256, 256, 0, stream>>>(xa, xb, perm2, 512);
    degree_rsqrt<<<1024, 256, 0, stream>>>(A2, dinv, 1024);
    xw_scale<<<(1024 * HID + 255) / 256, 256, 0, stream>>>(xa, uW0, dinv, Yp, 1024, HID);
    gcn_prop<<<1024 / 32, 128, 0, stream>>>(A2, Yp, dinv, ub0, xb, 1024, 1);

    // -------------------- up path i=1 (j=1, n=2048) ------------------------
    (void)hipMemcpyAsync(xa, xs1, (size_t)2048 * HID * sizeof(float),
                         hipMemcpyDeviceToDevice, stream);
    scatter_add<<<(1024 * HID + 255) / 256, 256, 0, stream>>>(xa, xb, perm1, 1024);
    degree_rsqrt<<<2048, 256, 0, stream>>>(A1, dinv, 2048);
    xw_scale<<<(2048 * HID + 255) / 256, 256, 0, stream>>>(xa, uW1, dinv, Yp, 2048, HID);
    gcn_prop<<<2048 / 32, 128, 0, stream>>>(A1, Yp, dinv, ub1, xb, 2048, 1);

    // -------------------- up path i=2 (j=0, n=4096, W 32x1) + sigmoid ------
    (void)hipMemcpyAsync(xa, xs0, (size_t)4096 * HID * sizeof(float),
                         hipMemcpyDeviceToDevice, stream);
    scatter_add<<<(2048 * HID + 255) / 256, 256, 0, stream>>>(xa, xb, perm0, 2048);
    degree_rsqrt<<<4096, 256, 0, stream>>>(A0, dinv, 4096);
    xw_scale_vec<<<(4096 + 255) / 256, 256, 0, stream>>>(xa, uW2, dinv, yvec, 4096);
    gcn_prop_vec_sigmoid<<<4096, 256, 0, stream>>>(A0, yvec, dinv, ub2,
                                                   (float*)d_out, 4096);
}